// ContinousNormalizingFlowRHS_13632226197801
// MI455X (gfx1250) — compile-verified
//
#include <hip/hip_runtime.h>
#include <math.h>

typedef __attribute__((ext_vector_type(2))) float v2f;
typedef __attribute__((ext_vector_type(8))) float v8f;

#define ZD   128
#define LDZ  129        // z_and_logpz / output row stride (Z+1)
#define FD   2048
#define ND   256
#define BD   8192
#define WOT_PITCH 20    // padded stride of transposed w_out tile (bank-conflict free b64 reads)

// ---------------------------------------------------------------------------
// Kernel 1: tiny hypernet.  h0 = tanh(W1*t + B1); h1 = tanh(W2 @ h0 + B2)
// ---------------------------------------------------------------------------
__global__ __launch_bounds__(256) void k_h1(
    const float* __restrict__ t,  const float* __restrict__ W1,
    const float* __restrict__ B1, const float* __restrict__ W2,
    const float* __restrict__ B2, float* __restrict__ h1)
{
    __shared__ float h0[4 * ND];
    const int tid = threadIdx.x;
    const float ts = t[0];
    #pragma unroll
    for (int k = 0; k < 4; ++k)
        h0[k * ND + tid] = tanhf(W1[k * ND + tid] * ts + B1[k * ND + tid]);
    __syncthreads();
    for (int k = 0; k < 4; ++k) {
        float acc = B2[k * ND + tid];
        const float* w = W2 + ((long)k * ND + tid) * ND;
        const float* x = h0 + k * ND;
        for (int m = 0; m < ND; ++m) acc += w[m] * x[m];
        h1[k * ND + tid] = tanhf(acc);
    }
}

// ---------------------------------------------------------------------------
// Kernel 2: hyper-weight matvec.  y[r] = W[r,:]·x + bias[r], row length 256.
// One wave32 per row, lane holds 8 contiguous floats (two B128 loads = the
// whole 1KB row coalesced).  Bandwidth-critical: 537 MB across two launches.
// ---------------------------------------------------------------------------
__global__ __launch_bounds__(256) void k_matvec(
    const float* __restrict__ W, const float* __restrict__ bias,
    const float* __restrict__ x, float* __restrict__ y, int rows)
{
    const int lane = threadIdx.x & 31;
    const int wpb  = blockDim.x >> 5;
    long wave = (long)blockIdx.x * wpb + (threadIdx.x >> 5);
    float xr[8];
    #pragma unroll
    for (int i = 0; i < 8; ++i) xr[i] = x[lane * 8 + i];

    for (long r = wave; r < rows; r += (long)gridDim.x * wpb) {
        const float* w = W + r * ND + lane * 8;
        float4 wa = *(const float4*)(w);
        float4 wb = *(const float4*)(w + 4);
        float acc = wa.x * xr[0] + wa.y * xr[1] + wa.z * xr[2] + wa.w * xr[3]
                  + wb.x * xr[4] + wb.y * xr[5] + wb.z * xr[6] + wb.w * xr[7];
        #pragma unroll
        for (int off = 16; off > 0; off >>= 1) acc += __shfl_xor(acc, off, 32);
        if (lane == 0) y[r] = acc + bias[r];
    }
}

// ---------------------------------------------------------------------------
// Kernel 3: per-feature head vectors (one wave per f):
//   b[f] = W3_b[f,:]·h1[2]+b3_b[f];  gate[f] = sigmoid(W3_g[f,:]·h1[3]+b3_g[f])
//   s[f] = w_in[f,:]·w_out[f,:]
// ---------------------------------------------------------------------------
__global__ __launch_bounds__(256) void k_head(
    const float* __restrict__ W3b, const float* __restrict__ b3b,
    const float* __restrict__ W3g, const float* __restrict__ b3g,
    const float* __restrict__ h1,
    const float* __restrict__ w_in, const float* __restrict__ w_out,
    float* __restrict__ bv, float* __restrict__ gv, float* __restrict__ sv)
{
    const int lane = threadIdx.x & 31;
    const int f    = blockIdx.x * (blockDim.x >> 5) + (threadIdx.x >> 5);
    const float* x2 = h1 + 2 * ND;
    const float* x3 = h1 + 3 * ND;

    float ab = 0.f, ag = 0.f, as = 0.f;
    const float* wb = W3b + (long)f * ND + lane * 8;
    const float* wg = W3g + (long)f * ND + lane * 8;
    #pragma unroll
    for (int i = 0; i < 8; ++i) {
        ab += wb[i] * x2[lane * 8 + i];
        ag += wg[i] * x3[lane * 8 + i];
    }
    const float* wi = w_in  + (long)f * ZD + lane * 4;
    const float* wo = w_out + (long)f * ZD + lane * 4;
    #pragma unroll
    for (int i = 0; i < 4; ++i) as += wi[i] * wo[i];

    #pragma unroll
    for (int off = 16; off > 0; off >>= 1) {
        ab += __shfl_xor(ab, off, 32);
        ag += __shfl_xor(ag, off, 32);
        as += __shfl_xor(as, off, 32);
    }
    if (lane == 0) {
        bv[f] = ab + b3b[f];
        gv[f] = 1.0f / (1.0f + expf(-(ag + b3g[f])));
        sv[f] = as;
    }
}

// ---------------------------------------------------------------------------
// Kernel 4: fused dual-GEMM + epilogue, fp32 WMMA (v_wmma_f32_16x16x4_f32).
// F-split-K: block (4 waves, 64 batch rows) owns an F-range of nft*16
// features; partial dz/trace go to `part` (reduced by k_reduce).
//   - w_in tile  (16x128) : GLOBAL_LOAD_ASYNC_TO_LDS_B128 (ASYNCcnt path)
//   - w_out tile : stored transposed [z][ff] (pitch 20) so GEMM2 B-fragments
//     are single aligned ds_load_b64 ops
//   - g tile transposed C-layout -> A-layout through per-wave LDS scratch
// ---------------------------------------------------------------------------
__global__ __launch_bounds__(128) void k_main(
    const float* __restrict__ zlog,
    const float* __restrict__ w_in, const float* __restrict__ w_out,
    const float* __restrict__ bv,   const float* __restrict__ gv,
    const float* __restrict__ sv,   float* __restrict__ part, int nft)
{
    __shared__ __align__(16) float s_win[16 * ZD];     // [ff][z]
    __shared__ float s_woutT[ZD * WOT_PITCH];          // [z][ff], padded
    __shared__ float s_b[16], s_g[16], s_s[16];
    __shared__ float s_G[4][16 * 16];                  // per-wave g transpose

    const int tid  = threadIdx.x;
    const int wave = tid >> 5;
    const int lane = tid & 31;
    const int m    = lane & 15;      // col (C/B layout) / row (A layout)
    const int hi   = lane >> 4;      // half-wave select
    const int bblk = blockIdx.x & 127;
    const int spl  = blockIdx.x >> 7;
    const int rb   = bblk * 64 + wave * 16;
    const int fbase = spl * nft * 16;

    // A fragments: z[rb..rb+15, 0..127], WMMA f32 16x16x4 A-layout:
    // lanes 0-15 hold K=4c+{0,1}; lanes 16-31 hold K=4c+{2,3}.
    v2f A[32];
    #pragma unroll
    for (int c = 0; c < 32; ++c) {
        const float* zp = zlog + (long)(rb + m) * LDZ + c * 4 + hi * 2;
        A[c].x = zp[0];
        A[c].y = zp[1];
    }

    v8f   dz[8] = {};
    float tr[8] = {};
    const unsigned lds_win = (unsigned)(uintptr_t)&s_win[0];
    const unsigned long long winBase = (unsigned long long)(uintptr_t)w_in;

    for (int ft = 0; ft < nft; ++ft) {
        const int f0 = fbase + ft * 16;
        __syncthreads();                       // LDS tiles free for reuse

        // w_in tile: 8KB contiguous copy via async-to-LDS (4 x B128 / thread)
        #pragma unroll
        for (int q = 0; q < 4; ++q) {
            unsigned off  = (unsigned)(q * 128 + tid) * 16u;   // bytes in tile
            unsigned ldsa = lds_win + off;
            unsigned goff = (unsigned)f0 * (unsigned)(ZD * 4) + off;
            asm volatile("global_load_async_to_lds_b128 %0, %1, %2"
                         :: "v"(ldsa), "v"(goff), "s"(winBase) : "memory");
        }
        // w_out tile: transpose into LDS ([z][ff], pitch 20)
        for (int i = tid; i < 16 * ZD; i += 128) {
            int ff = i >> 7, z = i & 127;
            s_woutT[z * WOT_PITCH + ff] = w_out[(long)f0 * ZD + i];
        }
        if (tid < 16) {
            s_b[tid] = bv[f0 + tid];
            s_g[tid] = gv[f0 + tid];
            s_s[tid] = sv[f0 + tid];
        }
        asm volatile("s_wait_asynccnt 0" ::: "memory");
        __syncthreads();

        // GEMM1: C1[16b x 16f] = z-tile(16x128) x w_in^T-tile(128x16)
        v8f C1 = {};
        #pragma unroll
        for (int c = 0; c < 32; ++c) {
            v2f Bf = *(const v2f*)&s_win[m * ZD + c * 4 + hi * 2]; // B[k][n]=w_in[f0+n][k]
            C1 = __builtin_amdgcn_wmma_f32_16x16x4_f32(
                     false, A[c], false, Bf, (short)0, C1, false, false);
        }

        // Epilogue: tanh/gate/trace; stage g for C->A layout transpose.
        {
            const float bn = s_b[m], gn = s_g[m], sn = s_s[m];
            #pragma unroll
            for (int r = 0; r < 8; ++r) {
                const float h = tanhf(C1[r] + bn);
                tr[r] += (1.0f - h * h) * gn * sn;
                s_G[wave][(r + hi * 8) * 16 + m] = h * gn;
            }
        }
        __syncthreads();                       // order stores before reads

        // g tile as A fragments (K=16 -> 4 chunks of 4)
        v2f Gf[4];
        #pragma unroll
        for (int c = 0; c < 4; ++c)
            Gf[c] = *(const v2f*)&s_G[wave][m * 16 + c * 4 + hi * 2];

        // GEMM2: dz(16x128) += g(16x16) x w_out-tile(16x128)
        #pragma unroll
        for (int j = 0; j < 8; ++j) {
            #pragma unroll
            for (int c = 0; c < 4; ++c) {
                const int k0 = c * 4 + hi * 2;
                v2f Bf = *(const v2f*)&s_woutT[(j * 16 + m) * WOT_PITCH + k0];
                dz[j] = __builtin_amdgcn_wmma_f32_16x16x4_f32(
                            false, Gf[c], false, Bf, (short)0, dz[j], false, false);
            }
        }
    }

    // Write raw partials: dz to cols 0..127, raw trace to col 128.
    float* po = part + (long)spl * BD * LDZ;
    #pragma unroll
    for (int j = 0; j < 8; ++j)
        #pragma unroll
        for (int r = 0; r < 8; ++r)
            po[(long)(rb + r + hi * 8) * LDZ + j * 16 + m] = dz[j][r];

    #pragma unroll
    for (int r = 0; r < 8; ++r) {
        float v = tr[r];
        #pragma unroll
        for (int off = 1; off < 16; off <<= 1) v += __shfl_xor(v, off, 32);
        tr[r] = v;                   // valid within each 16-lane half
    }
    if (m == 0) {                    // lane 0 -> rows 0..7, lane 16 -> 8..15
        #pragma unroll
        for (int r = 0; r < 8; ++r)
            po[(long)(rb + r + hi * 8) * LDZ + ZD] = tr[r];
    }
}

// ---------------------------------------------------------------------------
// Kernel 5: combine F-split partials; scale dz by 1/F, trace by -1/F.
// ---------------------------------------------------------------------------
__global__ __launch_bounds__(256) void k_reduce(
    const float* __restrict__ part, float* __restrict__ out, int n, int split)
{
    int i = blockIdx.x * blockDim.x + threadIdx.x;
    if (i >= n) return;
    float acc = 0.f;
    for (int s = 0; s < split; ++s) acc += part[(long)s * BD * LDZ + i];
    const float invF = 1.0f / (float)FD;
    const int col = i % LDZ;
    out[i] = (col == ZD) ? (-acc * invF) : (acc * invF);
}

// ---------------------------------------------------------------------------
extern "C" void kernel_launch(void* const* d_in, const int* in_sizes, int n_in,
                              void* d_out, int out_size, void* d_ws, size_t ws_size,
                              hipStream_t stream)
{
    const float* t       = (const float*)d_in[0];
    const float* zlog    = (const float*)d_in[1];
    const float* W1      = (const float*)d_in[2];
    const float* B1      = (const float*)d_in[3];
    const float* W2      = (const float*)d_in[4];
    const float* B2      = (const float*)d_in[5];
    const float* W3_win  = (const float*)d_in[6];
    const float* b3_win  = (const float*)d_in[7];
    const float* W3_wout = (const float*)d_in[8];
    const float* b3_wout = (const float*)d_in[9];
    const float* W3_b    = (const float*)d_in[10];
    const float* b3_b    = (const float*)d_in[11];
    const float* W3_g    = (const float*)d_in[12];
    const float* b3_g    = (const float*)d_in[13];
    float* out = (float*)d_out;

    // ws layout (floats): h1 | w_in | w_out | b | gate | s | partials
    float* ws    = (float*)d_ws;
    float* h1    = ws;
    float* w_in  = h1 + 4 * ND;
    float* w_out = w_in  + (long)FD * ZD;
    float* bv    = w_out + (long)FD * ZD;
    float* gv    = bv + FD;
    float* sv    = gv + FD;
    float* part  = sv + FD;

    const size_t baseF  = 4 * ND + 2L * FD * ZD + 3 * FD;
    const size_t availF = ws_size / sizeof(float);
    int split = 1;                                   // deterministic: ws_size fixed
    if (availF >= baseF + 4L * BD * LDZ) split = 4;
    else if (availF >= baseF + 2L * BD * LDZ) split = 2;

    const int R = FD * ZD;   // 262144 rows per hyper-weight matrix

    k_h1    <<<1, 256, 0, stream>>>(t, W1, B1, W2, B2, h1);
    k_matvec<<<R / 8, 256, 0, stream>>>(W3_win,  b3_win,  h1,      w_in,  R);
    k_matvec<<<R / 8, 256, 0, stream>>>(W3_wout, b3_wout, h1 + ND, w_out, R);
    k_head  <<<FD / 8, 256, 0, stream>>>(W3_b, b3_b, W3_g, b3_g, h1,
                                         w_in, w_out, bv, gv, sv);
    k_main  <<<128 * split, 128, 0, stream>>>(zlog, w_in, w_out, bv, gv, sv,
                                              part, (FD / 16) / split);
    k_reduce<<<(BD * LDZ + 255) / 256, 256, 0, stream>>>(part, out, BD * LDZ, split);
}